// TimesBlock_53197464928666
// MI455X (gfx1250) — compile-verified
//
#include <hip/hip_runtime.h>
#include <math.h>

#define B_    8
#define C_    64
#define L_    512
#define KNUM  4
#define CMAXI 32
#define PMAXI 512
#define KC    (3 * C_)   // 192 = fuse GEMM reduction dim

typedef __attribute__((ext_vector_type(2)))  float    v2f;
typedef __attribute__((ext_vector_type(8)))  float    v8f;

__device__ __forceinline__ float gelu_exact(float x) {
    return 0.5f * x * (1.0f + erff(x * 0.70710678118654752440f));
}

// ---------------------------------------------------------------------------
// Kernel 0: new_step_mask[b,p] = (p < max_k period_k) && (step_mask[b,p] > 0)
// Written to the tail of d_out (second tuple output).
// ---------------------------------------------------------------------------
__global__ void nsm_kernel(const float* __restrict__ step_mask,
                           const int* __restrict__ periods,
                           float* __restrict__ nsm_out) {
    int i = blockIdx.x * blockDim.x + threadIdx.x;
    if (i >= B_ * L_) return;
    int b = i / L_, p = i % L_;
    int maxP = 0;
    for (int k = 0; k < KNUM; ++k) {
        int P = periods[k];
        P = P < 16 ? 16 : (P > PMAXI ? PMAXI : P);
        maxP = P > maxP ? P : maxP;
    }
    nsm_out[b * L_ + p] = (p < maxP && step_mask[b * L_ + p] > 0.f) ? 1.f : 0.f;
}

// ---------------------------------------------------------------------------
// Kernel 1: per-(b,c) fold + inception conv + pooling + branch reconstruction.
// One 512-thread workgroup per (b,c); thread t owns output column p = t.
// The 128 MB fold is never materialized: conv_in[k,cy,p] is an LDS read of
// the 512-float feature row with a validity predicate.
// Writes combined TRANSPOSED as (B, L, 192) so kernel 2's B-operand lanes
// read contiguous K-pairs (global_load_b64).
// ---------------------------------------------------------------------------
__global__ __launch_bounds__(512) void fold_conv_kernel(
    const float* __restrict__ features,   // (B,C,L)
    const float* __restrict__ step_mask,  // (B,1,L)
    const float* __restrict__ amplitudes, // (K)
    const float* __restrict__ w0, const float* __restrict__ b0p, // (1,K,3,3)
    const float* __restrict__ w1, const float* __restrict__ b1p, // (1,K,5,5)
    const float* __restrict__ w2, const float* __restrict__ b2p, // (1,K,7,7)
    const float* __restrict__ wproj, const float* __restrict__ bproj, // (1,3,1,1)
    const float* __restrict__ wres,  const float* __restrict__ bres,  // (1,K,1,1)
    const int*   __restrict__ periods,
    float* __restrict__ comb_t)           // (B, L, 192) scratch
{
    __shared__ float feat_s[L_];
    __shared__ float w0s[KNUM * 9];
    __shared__ float w1s[KNUM * 25];
    __shared__ float w2s[KNUM * 49];
    __shared__ float wres_s[KNUM], amp_s[KNUM], wproj_s[3];
    __shared__ float bias_s[5]; // b0,b1,b2,bproj,bres

    const int bc  = blockIdx.x;
    const int b   = bc / C_;
    const int c   = bc % C_;
    const int tid = threadIdx.x;

    if (tid < KNUM * 9)  w0s[tid] = w0[tid];
    if (tid < KNUM * 25) w1s[tid] = w1[tid];
    if (tid < KNUM * 49) w2s[tid] = w2[tid];
    if (tid < KNUM) { wres_s[tid] = wres[tid]; amp_s[tid] = amplitudes[tid]; }
    if (tid < 3)    wproj_s[tid] = wproj[tid];
    if (tid == 0) {
        bias_s[0] = b0p[0]; bias_s[1] = b1p[0]; bias_s[2] = b2p[0];
        bias_s[3] = bproj[0]; bias_s[4] = bres[0];
    }

    const float sm    = step_mask[b * L_ + tid];
    const float f_raw = features[((size_t)b * C_ + c) * L_ + tid];
    feat_s[tid] = f_raw * sm;   // features_in
    __syncthreads();

    int Pk[KNUM], Ck[KNUM], Bk[KNUM];
    int maxP = 0;
#pragma unroll
    for (int k = 0; k < KNUM; ++k) {
        int P = periods[k];
        P = P < 16 ? 16 : (P > PMAXI ? PMAXI : P);
        Pk[k] = P; Ck[k] = L_ / P; Bk[k] = L_ - Ck[k] * P;
        maxP = P > maxP ? P : maxP;
    }

    auto cin = [&](int k, int cy, int pp) -> float {
        if (cy < 0 || pp < 0 || cy >= Ck[k] || pp >= Pk[k]) return 0.f;
        return feat_s[Bk[k] + cy * Pk[k] + pp];
    };

    const int   p   = tid;
    const float eps = 1.1920929e-7f;           // finfo(f32).eps
    const float NEG = -3.4028234663852886e38f; // finfo(f32).min

    float zsum = 0.f, zmax = NEG;
    int   wsum = 0;

#pragma unroll 1
    for (int cy = 0; cy < CMAXI; ++cy) {
        bool anyv = false;
#pragma unroll
        for (int k = 0; k < KNUM; ++k) anyv |= (cy < Ck[k] && p < Pk[k]);

        float f0 = bias_s[0], f1 = bias_s[1], f2 = bias_s[2];
        float res = bias_s[4];
#pragma unroll
        for (int k = 0; k < KNUM; ++k) res += wres_s[k] * cin(k, cy, p);

#pragma unroll 1
        for (int dy = -3; dy <= 3; ++dy) {
#pragma unroll 1
            for (int dx = -3; dx <= 3; ++dx) {
                const int ady = dy < 0 ? -dy : dy;
                const int adx = dx < 0 ? -dx : dx;
#pragma unroll
                for (int k = 0; k < KNUM; ++k) {
                    const float v = cin(k, cy + dy, p + dx);
                    f2 += w2s[k * 49 + (dy + 3) * 7 + (dx + 3)] * v;
                    if (ady <= 2 && adx <= 2)
                        f1 += w1s[k * 25 + (dy + 2) * 5 + (dx + 2)] * v;
                    if (ady <= 1 && adx <= 1)
                        f0 += w0s[k * 9 + (dy + 1) * 3 + (dx + 1)] * v;
                }
            }
        }
        float z = wproj_s[0] * f0 + wproj_s[1] * f1 + wproj_s[2] * f2 + bias_s[3];
        z = gelu_exact(z) + res;
        if (anyv) { zsum += z; wsum += 1; zmax = fmaxf(zmax, z); }
    }

    const float avgp = zsum / ((float)wsum + eps);
    const float maxp = (wsum > 0) ? zmax : 0.f;

    // branch softmax over amplitudes (all branches valid: cycles>=1, period>=16)
    float mamp = amp_s[0];
#pragma unroll
    for (int k = 1; k < KNUM; ++k) mamp = fmaxf(mamp, amp_s[k]);
    float bwk[KNUM], ssum = 0.f;
#pragma unroll
    for (int k = 0; k < KNUM; ++k) { bwk[k] = expf(amp_s[k] - mamp); ssum += bwk[k]; }
    float denom = 0.f;
#pragma unroll
    for (int k = 0; k < KNUM; ++k) { bwk[k] /= ssum; denom += bwk[k]; }

    float wrec = 0.f;
#pragma unroll 1
    for (int k = 0; k < KNUM; ++k) {
        const float wsafe = (denom > 0.f) ? bwk[k] / (denom + eps) : 0.f;
        float brec = 0.f;
        if (p < Pk[k]) {
            float bs = 0.f;
            for (int cy = 0; cy < Ck[k]; ++cy) bs += feat_s[Bk[k] + cy * Pk[k] + p];
            brec = bs / ((float)Ck[k] + eps);
        }
        wrec += brec * wsafe;
    }

    const float nsm  = (p < maxP && sm > 0.f) ? 1.f : 0.f;
    const float fmod = (f_raw + wrec * nsm) * nsm;

    // transposed combined: row = column p, 192 channels contiguous
    float* row = comb_t + ((size_t)b * L_ + p) * KC;
    row[c]          = avgp * nsm;   // avg_pool
    row[C_ + c]     = maxp * nsm;   // max_pool
    row[2 * C_ + c] = fmod;         // features_mod
}

// ---------------------------------------------------------------------------
// Kernel 2: fuse GEMM  update[b,o,l] = gelu(sum_c w[o,c]*combined[b,c,l]+bias)*nsm
// M=64, N=4096, K=192.  V_WMMA_F32_16X16X4_F32, 48 k-steps per tile.
// Block = one (b, o-tile): A tile (16x192 = 12 KB) staged in LDS once.
// Each of 8 waves owns 4 l-tiles -> 4 independent accumulators reusing the
// same A fragment; k-loop unrolled 4x so loads clause up and 16 WMMAs are in
// flight per unrolled body.
// ---------------------------------------------------------------------------
__global__ __launch_bounds__(256) void fuse_gemm_kernel(
    const float* __restrict__ comb_t, // (B, L, 192)
    const float* __restrict__ wfuse,  // (64, 192)
    const float* __restrict__ bfuse,  // (64)
    const float* __restrict__ nsm,    // (B, L)
    float* __restrict__ out)          // (B, 64, L)
{
    __shared__ float wA[16 * KC];     // 12 KB A tile

    const int blk  = blockIdx.x;      // 0..31 : b*4 + o-tile
    const int b    = blk >> 2;
    const int o0   = (blk & 3) * 16;
    const int tid  = threadIdx.x;
    const int wid  = tid >> 5;        // 8 waves
    const int lane = tid & 31;
    const bool hi  = lane >= 16;
    const int  lm  = lane & 15;

    // cooperative stage of A tile: rows o0..o0+15 of wfuse
    for (int i = tid; i < 16 * KC; i += 256)
        wA[i] = wfuse[(o0 + (i / KC)) * KC + (i % KC)];
    __syncthreads();

    const float* cbase = comb_t + (size_t)b * L_ * KC;

    // 4 l-tiles per wave; per-lane B row pointers (contiguous K in memory)
    const int l0 = wid * 64;          // this wave covers l0 .. l0+63
    const float* br0 = cbase + (size_t)(l0 +  0 + lm) * KC;
    const float* br1 = cbase + (size_t)(l0 + 16 + lm) * KC;
    const float* br2 = cbase + (size_t)(l0 + 32 + lm) * KC;
    const float* br3 = cbase + (size_t)(l0 + 48 + lm) * KC;

    v8f acc0 = {}, acc1 = {}, acc2 = {}, acc3 = {};

    // A 16x4 layout: lanes 0-15 -> K={kk,kk+1}, lanes 16-31 -> K={kk+2,kk+3}
    // B 4x16 layout mirrored: N = lane%16, same K split.
    const int koff = hi ? 2 : 0;
    const float* arow = &wA[lm * KC];

#pragma unroll 4
    for (int kk = 0; kk < KC; kk += 4) {
        const int ka = kk + koff;
        const v2f a  = *(const v2f*)&arow[ka];   // ds_load_b64
        const v2f b0 = *(const v2f*)&br0[ka];    // global_load_b64 x4
        const v2f b1 = *(const v2f*)&br1[ka];
        const v2f b2 = *(const v2f*)&br2[ka];
        const v2f b3 = *(const v2f*)&br3[ka];
        acc0 = __builtin_amdgcn_wmma_f32_16x16x4_f32(false, a, false, b0, (short)0, acc0, false, false);
        acc1 = __builtin_amdgcn_wmma_f32_16x16x4_f32(false, a, false, b1, (short)0, acc1, false, false);
        acc2 = __builtin_amdgcn_wmma_f32_16x16x4_f32(false, a, false, b2, (short)0, acc2, false, false);
        acc3 = __builtin_amdgcn_wmma_f32_16x16x4_f32(false, a, false, b3, (short)0, acc3, false, false);
    }

    // epilogue: bias + exact gelu + mask, one 16x16 tile per accumulator
    v8f accs[4] = {acc0, acc1, acc2, acc3};
#pragma unroll
    for (int t = 0; t < 4; ++t) {
        const int lcol = l0 + t * 16 + lm;
        const float nv = nsm[b * L_ + lcol];
#pragma unroll
        for (int g = 0; g < 8; ++g) {
            const int M = o0 + g + (hi ? 8 : 0);
            float v = accs[t][g] + bfuse[M];
            v = gelu_exact(v) * nv;
            out[((size_t)b * C_ + M) * L_ + lcol] = v;
        }
    }
}

// ---------------------------------------------------------------------------
extern "C" void kernel_launch(void* const* d_in, const int* in_sizes, int n_in,
                              void* d_out, int out_size, void* d_ws, size_t ws_size,
                              hipStream_t stream)
{
    (void)in_sizes; (void)n_in; (void)out_size; (void)ws_size;

    const float* features   = (const float*)d_in[0];
    const float* step_mask  = (const float*)d_in[1];
    const float* amplitudes = (const float*)d_in[2];
    const float* w0    = (const float*)d_in[3];
    const float* b0    = (const float*)d_in[4];
    const float* w1    = (const float*)d_in[5];
    const float* b1    = (const float*)d_in[6];
    const float* w2    = (const float*)d_in[7];
    const float* b2    = (const float*)d_in[8];
    const float* wproj = (const float*)d_in[9];
    const float* bproj = (const float*)d_in[10];
    const float* wres  = (const float*)d_in[11];
    const float* bres  = (const float*)d_in[12];
    const float* wfuse = (const float*)d_in[13];
    const float* bfuse = (const float*)d_in[14];
    const int*   periods = (const int*)d_in[15];

    float* out_update = (float*)d_out;                       // (B,64,L)
    float* out_nsm    = out_update + (size_t)B_ * C_ * L_;   // (B,1,L)
    float* comb_t     = (float*)d_ws;                        // (B,L,192) = 3 MB

    nsm_kernel<<<(B_ * L_ + 255) / 256, 256, 0, stream>>>(step_mask, periods, out_nsm);

    fold_conv_kernel<<<B_ * C_, 512, 0, stream>>>(
        features, step_mask, amplitudes,
        w0, b0, w1, b1, w2, b2, wproj, bproj, wres, bres,
        periods, comb_t);

    fuse_gemm_kernel<<<32, 256, 0, stream>>>(comb_t, wfuse, bfuse, out_nsm, out_update);
}